// TFN_61143154426197
// MI455X (gfx1250) — compile-verified
//
#include <hip/hip_runtime.h>
#include <math.h>

typedef __attribute__((ext_vector_type(2))) float v2f;
typedef __attribute__((ext_vector_type(8))) float v8f;

#define GSCALE (0.69314718056f * 9.0f)

// ---------------------------------------------------------------------------
// Spherical harmonics, degrees 0..3 (16 values)
// ---------------------------------------------------------------------------
__device__ __forceinline__ void sh16(float x, float y, float z, float* Y) {
  Y[0]  = 0.28209479f;
  Y[1]  = 0.48860251f * y;
  Y[2]  = 0.48860251f * z;
  Y[3]  = 0.48860251f * x;
  Y[4]  = 1.09254843f * x * y;
  Y[5]  = 1.09254843f * y * z;
  Y[6]  = 0.31539157f * (3.f * z * z - 1.f);
  Y[7]  = 1.09254843f * x * z;
  Y[8]  = 0.54627421f * (x * x - y * y);
  Y[9]  = 0.59004359f * y * (3.f * x * x - y * y);
  Y[10] = 2.89061144f * x * y * z;
  Y[11] = 0.45704580f * y * (5.f * z * z - 1.f);
  Y[12] = 0.37317633f * z * (5.f * z * z - 3.f);
  Y[13] = 0.45704580f * x * (5.f * z * z - 1.f);
  Y[14] = 1.44530572f * z * (x * x - y * y);
  Y[15] = 0.59004359f * x * (x * x - 3.f * y * y);
}

// ---------------------------------------------------------------------------
// kd-tree level sort: one workgroup per block of m points, bitonic sort in LDS
// along the largest-extent axis.
// ---------------------------------------------------------------------------
__global__ void kd_sort_kernel(const float* __restrict__ in, float* __restrict__ out, int m) {
  __shared__ float sx[1024], sy[1024], sz[1024], key[1024];
  __shared__ int   pid[1024];
  __shared__ float sPart[256][6];
  __shared__ int   sAx;
  int tid = threadIdx.x;
  long base = (long)blockIdx.x * m * 3;

  float mnx = 3.4e38f, mny = 3.4e38f, mnz = 3.4e38f;
  float mxx = -3.4e38f, mxy = -3.4e38f, mxz = -3.4e38f;
  for (int i = tid; i < m; i += blockDim.x) {
    float a = in[base + 3 * i + 0];
    float b = in[base + 3 * i + 1];
    float c = in[base + 3 * i + 2];
    sx[i] = a; sy[i] = b; sz[i] = c;
    mnx = fminf(mnx, a); mny = fminf(mny, b); mnz = fminf(mnz, c);
    mxx = fmaxf(mxx, a); mxy = fmaxf(mxy, b); mxz = fmaxf(mxz, c);
  }
  sPart[tid][0] = mnx; sPart[tid][1] = mny; sPart[tid][2] = mnz;
  sPart[tid][3] = mxx; sPart[tid][4] = mxy; sPart[tid][5] = mxz;
  __syncthreads();
  if (tid == 0) {
    float n0 = 3.4e38f, n1 = 3.4e38f, n2 = 3.4e38f;
    float x0 = -3.4e38f, x1 = -3.4e38f, x2 = -3.4e38f;
    for (int t = 0; t < 256; t++) {
      n0 = fminf(n0, sPart[t][0]); n1 = fminf(n1, sPart[t][1]); n2 = fminf(n2, sPart[t][2]);
      x0 = fmaxf(x0, sPart[t][3]); x1 = fmaxf(x1, sPart[t][4]); x2 = fmaxf(x2, sPart[t][5]);
    }
    float e0 = x0 - n0, e1 = x1 - n1, e2 = x2 - n2;
    int ax = 0; float eb = e0;
    if (e1 > eb) { ax = 1; eb = e1; }
    if (e2 > eb) { ax = 2; }
    sAx = ax;
  }
  __syncthreads();
  int ax = sAx;
  for (int i = tid; i < m; i += blockDim.x) {
    key[i] = (ax == 0) ? sx[i] : (ax == 1) ? sy[i] : sz[i];
    pid[i] = i;
  }
  __syncthreads();
  for (int size = 2; size <= m; size <<= 1) {
    for (int stride = size >> 1; stride > 0; stride >>= 1) {
      for (int i = tid; i < m; i += blockDim.x) {
        int p = i ^ stride;
        if (p > i) {
          bool up = ((i & size) == 0);
          float ki = key[i], kp = key[p];
          if ((ki > kp) == up) {
            key[i] = kp; key[p] = ki;
            int t = pid[i]; pid[i] = pid[p]; pid[p] = t;
          }
        }
      }
      __syncthreads();
    }
  }
  for (int i = tid; i < m; i += blockDim.x) {
    int j = pid[i];
    out[base + 3 * i + 0] = sx[j];
    out[base + 3 * i + 1] = sy[j];
    out[base + 3 * i + 2] = sz[j];
  }
}

// ---------------------------------------------------------------------------
// kd_pool: mean over stride consecutive points
// ---------------------------------------------------------------------------
__global__ void kd_pool_kernel(const float* __restrict__ in, float* __restrict__ out,
                               int Nout, int stride) {
  int i = blockIdx.x * blockDim.x + threadIdx.x;
  int tot = 32 * Nout * 3;
  if (i >= tot) return;
  int c = i % 3; int bn = i / 3; int n = bn % Nout; int b = bn / Nout;
  int Nin = Nout * stride;
  float s = 0.f;
  for (int j = 0; j < stride; j++) s += in[((long)b * Nin + n * stride + j) * 3 + c];
  out[i] = s / (float)stride;
}

// ---------------------------------------------------------------------------
// exact 32-NN per target point: one wave32 per (b,m); iterated wave-min with
// index tiebreak (matches lax.top_k ordering on -d2).
// ---------------------------------------------------------------------------
__global__ void knn_kernel(const float* __restrict__ src, const float* __restrict__ tgt,
                           int* __restrict__ idxOut, int N, int M) {
  int bm = blockIdx.x;
  int b = bm / M;
  int lane = threadIdx.x;
  float tx = tgt[bm * 3 + 0], ty = tgt[bm * 3 + 1], tz = tgt[bm * 3 + 2];
  int cnt = N >> 5;
  const float* sb = src + (long)b * N * 3;
  float d2[32];
  for (int t = 0; t < cnt; t++) {
    int g = t * 32 + lane;
    float dx = sb[g * 3 + 0] - tx;
    float dy = sb[g * 3 + 1] - ty;
    float dz = sb[g * 3 + 2] - tz;
    d2[t] = dx * dx + dy * dy + dz * dz;
  }
  unsigned used = 0u;
  for (int sel = 0; sel < 32; sel++) {
    float bv = 3.4e38f; int bg = 0x7fffffff;
    for (int t = 0; t < cnt; t++) {
      if (!((used >> t) & 1u)) {
        float v = d2[t]; int g = t * 32 + lane;
        if (v < bv || (v == bv && g < bg)) { bv = v; bg = g; }
      }
    }
    for (int off = 16; off > 0; off >>= 1) {
      float ov = __shfl_xor(bv, off, 32);
      int   og = __shfl_xor(bg, off, 32);
      if (ov < bv || (ov == bv && og < bg)) { bv = ov; bg = og; }
    }
    if ((bg & 31) == lane) used |= 1u << (bg >> 5);
    if (lane == sel) idxOut[(long)bm * 32 + sel] = bg;
  }
}

// ---------------------------------------------------------------------------
// Fused SH-Gaussian convolution for one level. One block per target point.
// ---------------------------------------------------------------------------
template <int LVL>
__global__ void sh_conv_kernel(const float* __restrict__ src, const float* __restrict__ tgt,
                               const int* __restrict__ idx,
                               const float* __restrict__ y0, const float* __restrict__ y1,
                               const float* __restrict__ y2, const float* __restrict__ y3,
                               float* __restrict__ o0, float* __restrict__ o1,
                               float* __restrict__ o2, float* __restrict__ o3) {
  constexpr int   N   = (LVL == 0) ? 1024 : (LVL == 1) ? 256 : 64;
  constexpr int   M   = (LVL == 0) ? 256 : (LVL == 1) ? 64 : 16;
  constexpr float RAD = (LVL == 0) ? 0.2f : (LVL == 1) ? 0.4f : 0.8f;
  constexpr int   C0  = (LVL == 0) ? 1 : (LVL == 1) ? 32 : 64;
  constexpr int   C1  = (LVL == 0) ? 1 : (LVL == 1) ? 33 : 65;
  constexpr int   C2  = (LVL == 0) ? 0 : (LVL == 1) ? 32 : 64;
  constexpr int   C3  = C2;
  constexpr int   O0  = 3 * (C0 + C1 + C2 + C3);
  constexpr int   O1c = 3 * (C0 + C1);
  constexpr int   O2  = 3 * (C0 + C2);
  constexpr int   O3  = 3 * (C0 + C3);
  constexpr int   E   = O0 + 3 * O1c + 5 * O2 + 7 * O3;

  __shared__ int   sIdx[32];
  __shared__ float sY[32 * 16];
  __shared__ float sG[32 * 3];
  __shared__ float sSrcYZX[32 * 3];
  __shared__ float sTgtYZX[3];

  int bm = blockIdx.x;
  int b  = bm / M;
  int tid = threadIdx.x;

  if (tid < 32) {
    int k = tid;
    int j = idx[(long)bm * 32 + k];
    sIdx[k] = j;
    const float* sp = src + ((long)b * N + j) * 3;
    const float* tp = tgt + (long)bm * 3;
    float px = sp[0] - tp[0], py = sp[1] - tp[1], pz = sp[2] - tp[2];
    float d2r = px * px + py * py + pz * pz;
    float dist = sqrtf(fmaxf(d2r, 1e-12f)) / RAD;
    px /= RAD; py /= RAD; pz /= RAD;
    float inv = 1.f / fmaxf(dist, 1e-8f);
    float Yv[16];
    sh16(px * inv, py * inv, pz * inv, Yv);
    for (int q = 0; q < 16; q++) sY[k * 16 + q] = Yv[q];
    float g0 = __expf(-GSCALE * (dist - 0.0f) * (dist - 0.0f));
    float g1 = __expf(-GSCALE * (dist - 0.5f) * (dist - 0.5f));
    float g2 = __expf(-GSCALE * (dist - 1.0f) * (dist - 1.0f));
    float gs = g0 + g1 + g2 + 1e-8f;
    float bnd = (dist <= 1.f) ? 1.f : 0.f;
    sG[k * 3 + 0] = g0 / gs * bnd;
    sG[k * 3 + 1] = g1 / gs * bnd;
    sG[k * 3 + 2] = g2 / gs * bnd;
    sSrcYZX[k * 3 + 0] = sp[1];
    sSrcYZX[k * 3 + 1] = sp[2];
    sSrcYZX[k * 3 + 2] = sp[0];
    if (k == 0) { sTgtYZX[0] = tp[1]; sTgtYZX[1] = tp[2]; sTgtYZX[2] = tp[0]; }
  }
  __syncthreads();

  auto feat0 = [&](int k, int c) -> float {
    if constexpr (LVL == 0) { (void)k; (void)c; return 1.f; }
    else return y0[((long)b * N + sIdx[k]) * C0 + c];
  };
  auto featL = [&](int l, int k, int u, int c) -> float {
    if (l == 1) {
      if constexpr (LVL == 0) { return sSrcYZX[k * 3 + u]; }
      else {
        if (c == C1 - 1) return sSrcYZX[k * 3 + u];
        return y1[(((long)b * N + sIdx[k]) * 3 + u) * (C1 - 1) + c];
      }
    } else if (l == 2) {
      return y2[(((long)b * N + sIdx[k]) * 5 + u) * (C2 > 0 ? C2 : 1) + c];
    } else {
      return y3[(((long)b * N + sIdx[k]) * 7 + u) * (C3 > 0 ? C3 : 1) + c];
    }
  };

  const int loffs[4] = {0, 1, 4, 9};

  for (int e = tid; e < E; e += blockDim.x) {
    int lo, v, ch, r = e;
    if (r < O0) { lo = 0; v = 0; ch = r; }
    else {
      r -= O0;
      if (r < 3 * O1c) { lo = 1; v = r / O1c; ch = r - v * O1c; }
      else {
        r -= 3 * O1c;
        if (r < 5 * O2) { lo = 2; v = r / O2; ch = r - v * O2; }
        else { r -= 5 * O2; lo = 3; v = r / O3; ch = r - v * O3; }
      }
    }
    float acc = 0.f;
    if (ch < 3 * C0) {
      int s = ch / C0, c = ch - s * C0;
      int yb = loffs[lo] + v;
      for (int k = 0; k < 32; k++) acc += sY[k * 16 + yb] * sG[k * 3 + s] * feat0(k, c);
    } else if (lo == 0) {
      int rr = ch - 3 * C0;
      int l = 1, s = 0, c = 0;
      if (rr < 3 * C1) { l = 1; s = rr / C1; c = rr - s * C1; }
      else {
        rr -= 3 * C1;
        if constexpr (C2 > 0) {
          if (rr < 3 * C2) { l = 2; s = rr / C2; c = rr - s * C2; }
          else { rr -= 3 * C2; l = 3; s = rr / C3; c = rr - s * C3; }
        }
      }
      int nb = 2 * l + 1, yb = loffs[l];
      for (int k = 0; k < 32; k++) {
        float g = sG[k * 3 + s];
        for (int u = 0; u < nb; u++) acc += sY[k * 16 + yb + u] * g * featL(l, k, u, c);
      }
    } else {
      int Cl = (lo == 1) ? C1 : (lo == 2) ? C2 : C3;
      if (Cl == 0) Cl = 1;
      int rr = ch - 3 * C0;
      int s = rr / Cl, c = rr - s * Cl;
      for (int k = 0; k < 32; k++) acc += 0.28209479f * sG[k * 3 + s] * featL(lo, k, v, c);
    }
    if (lo == 0)      o0[(long)bm * O0 + ch] = acc;
    else if (lo == 1) o1[((long)bm * 3 + v) * (O1c + 1) + ch] = acc;
    else if (lo == 2) o2[((long)bm * 5 + v) * O2 + ch] = acc;
    else              o3[((long)bm * 7 + v) * O3 + ch] = acc;
  }
  if (tid < 3) o1[((long)bm * 3 + tid) * (O1c + 1) + O1c] = sTgtYZX[tid];
}

// ---------------------------------------------------------------------------
// inv_norms for one degree: n[bm, off+c] = sqrt(max(sum_v y^2, 1e-8))
// ---------------------------------------------------------------------------
__global__ void inv_norm_kernel(const float* __restrict__ y, float* __restrict__ n,
                                int BM, int V, int C, int GS, int off) {
  int i = blockIdx.x * blockDim.x + threadIdx.x;
  if (i >= BM * C) return;
  int bm = i / C, c = i - bm * C;
  float s = 0.f;
  for (int v = 0; v < V; v++) {
    float t = y[((long)bm * V + v) * C + c];
    s += t * t;
  }
  n[(long)bm * GS + off + c] = sqrtf(fmaxf(s, 1e-8f));
}

// ---------------------------------------------------------------------------
// WMMA f32 GEMM: C[M,N] = A[M,K] * B[K,N], row-major.
// One wave per 16 x (TPW*16) strip; TPW accumulators share one A fragment per
// k-step. Running per-tile B pointers (+= 4N) keep 64-bit address math out of
// the loop. MASK=false (N multiple of strip width) has zero masking: loads
// feed v_wmma_f32_16x16x4_f32 directly. MASK=true clamps addresses and
// multiplies a 0/1 mask — still branch-free. M must be a multiple of 16.
// ---------------------------------------------------------------------------
template <int TPW, bool MASK>
__global__ void gemm_wmma_kernel(const float* __restrict__ A, const float* __restrict__ B,
                                 float* __restrict__ C, int M, int K, int N) {
  int strip = TPW * 16;
  int tilesNW = MASK ? ((N + strip - 1) / strip) : (N / strip);
  int tm = blockIdx.x / tilesNW;
  int tq = blockIdx.x - tm * tilesNW;
  int lane = threadIdx.x;
  int mrow = tm * 16 + (lane & 15);
  int kh = (lane >> 4) << 1;   // lanes 0-15 hold K {0,1}; lanes 16-31 hold {2,3}

  int colo[TPW], col[TPW];
  float cm[TPW];
#pragma unroll
  for (int t = 0; t < TPW; t++) {
    int c = tq * strip + t * 16 + (lane & 15);
    colo[t] = c;
    if (MASK) {
      col[t] = (c < N) ? c : (N - 1);
      cm[t]  = (c < N) ? 1.f : 0.f;
    } else {
      col[t] = c;
      cm[t]  = 1.f;
    }
  }

  v8f acc[TPW];
#pragma unroll
  for (int t = 0; t < TPW; t++) acc[t] = (v8f){0.f, 0.f, 0.f, 0.f, 0.f, 0.f, 0.f, 0.f};

  const float* ap = A + (long)mrow * K + kh;
  const float* bp[TPW];
#pragma unroll
  for (int t = 0; t < TPW; t++) bp[t] = B + (long)kh * N + col[t];
  long bstep = 4L * N;
  int K4 = K & ~3;

  for (int kk = 0; kk < K4; kk += 4) {
    v2f a;
    a.x = ap[0];
    a.y = ap[1];
    ap += 4;
#pragma unroll
    for (int t = 0; t < TPW; t++) {
      v2f bf;
      bf.x = bp[t][0];
      bf.y = bp[t][N];
      if (MASK) { bf.x *= cm[t]; bf.y *= cm[t]; }
      acc[t] = __builtin_amdgcn_wmma_f32_16x16x4_f32(false, a, false, bf, (short)0,
                                                     acc[t], false, false);
      bp[t] += bstep;
    }
  }

  if (K4 < K) {
    int k0 = K4 + kh, k1 = k0 + 1;
    int k0c = (k0 < K) ? k0 : (K - 1);
    int k1c = (k1 < K) ? k1 : (K - 1);
    float am0 = (k0 < K) ? 1.f : 0.f;
    float am1 = (k1 < K) ? 1.f : 0.f;
    v2f a;
    a.x = A[(long)mrow * K + k0c] * am0;   // zeroed A lanes kill out-of-range K
    a.y = A[(long)mrow * K + k1c] * am1;
    const float* B0 = B + (long)k0c * N;
    const float* B1 = B + (long)k1c * N;
#pragma unroll
    for (int t = 0; t < TPW; t++) {
      v2f bf;
      bf.x = B0[col[t]];
      bf.y = B1[col[t]];
      if (MASK) { bf.x *= cm[t]; bf.y *= cm[t]; }
      acc[t] = __builtin_amdgcn_wmma_f32_16x16x4_f32(false, a, false, bf, (short)0,
                                                     acc[t], false, false);
    }
  }

  int rb = tm * 16 + ((lane < 16) ? 0 : 8);
#pragma unroll
  for (int t = 0; t < TPW; t++) {
    if (!MASK || colo[t] < N) {
#pragma unroll
      for (int j = 0; j < 8; j++) C[(long)(rb + j) * N + colo[t]] = acc[t][j];
    }
  }
  (void)M;
}

// ---------------------------------------------------------------------------
// Epilogues
// ---------------------------------------------------------------------------
__global__ void relu_kernel(float* __restrict__ p, int n) {
  int i = blockIdx.x * blockDim.x + threadIdx.x;
  if (i < n) p[i] = fmaxf(p[i], 0.f);
}

__global__ void gate_apply_kernel(float* __restrict__ z, const float* __restrict__ gz,
                                  const float* __restrict__ gb, int BM, int V, int U) {
  long n = (long)BM * V * U;
  long i = (long)blockIdx.x * blockDim.x + threadIdx.x;
  if (i >= n) return;
  int u = (int)(i % U);
  long bmv = i / U;
  long bm = bmv / V;
  float g = gz[bm * U + u] + gb[u];
  g = 1.f / (1.f + __expf(-g));
  z[i] *= g;
}

__global__ void bias_act_kernel(float* __restrict__ p, const float* __restrict__ bias,
                                int rows, int cols, int doRelu) {
  int i = blockIdx.x * blockDim.x + threadIdx.x;
  if (i >= rows * cols) return;
  int c = i % cols;
  float v = p[i] + bias[c];
  if (doRelu) v = fmaxf(v, 0.f);
  p[i] = v;
}

__global__ void maxpool_kernel(const float* __restrict__ n3, float* __restrict__ f) {
  int i = blockIdx.x * blockDim.x + threadIdx.x;
  if (i >= 32 * 512) return;
  int b = i / 512, c = i - b * 512;
  float m = -3.4e38f;
  for (int p = 0; p < 16; p++) m = fmaxf(m, n3[((long)b * 16 + p) * 512 + c]);
  f[i] = m;
}

__global__ void softmax_kernel(const float* __restrict__ logits, const float* __restrict__ ob,
                               float* __restrict__ out) {
  int r = threadIdx.x;
  if (r >= 32) return;
  float v[40];
  float mx = -3.4e38f;
  for (int j = 0; j < 40; j++) { v[j] = logits[r * 40 + j] + ob[j]; mx = fmaxf(mx, v[j]); }
  float s = 0.f;
  for (int j = 0; j < 40; j++) { v[j] = __expf(v[j] - mx); s += v[j]; }
  for (int j = 0; j < 40; j++) out[r * 40 + j] = v[j] / s;
}

// ---------------------------------------------------------------------------
// Host orchestration
// ---------------------------------------------------------------------------
extern "C" void kernel_launch(void* const* d_in, const int* in_sizes, int n_in,
                              void* d_out, int out_size, void* d_ws, size_t ws_size,
                              hipStream_t stream) {
  (void)in_sizes; (void)n_in; (void)out_size; (void)ws_size;

  const float* X = (const float*)d_in[0];
  auto EQW = [&](int i, int l) { return (const float*)d_in[1 + i * 4 + l]; };
  auto GW  = [&](int i, int j) { return (const float*)d_in[13 + i * 3 + j]; };
  auto GB  = [&](int i, int j) { return (const float*)d_in[22 + i * 3 + j]; };
  const float* fc1w = (const float*)d_in[31];
  const float* fc1b = (const float*)d_in[32];
  const float* fc2w = (const float*)d_in[33];
  const float* fc2b = (const float*)d_in[34];
  const float* outw = (const float*)d_in[35];
  const float* outb = (const float*)d_in[36];
  float* OUT = (float*)d_out;

  // bump allocator over d_ws
  size_t off = 0;
  auto allocF = [&](size_t n) -> float* {
    float* p = (float*)((char*)d_ws + off);
    off = (off + n * sizeof(float) + 255) & ~(size_t)255;
    return p;
  };
  auto allocI = [&](size_t n) -> int* { return (int*)allocF(n); };

  float* xsA = allocF(32 * 1024 * 3);
  float* xsB = allocF(32 * 1024 * 3);
  float* pts1 = allocF(32 * 256 * 3);
  float* pts2 = allocF(32 * 64 * 3);
  float* pts3 = allocF(32 * 16 * 3);
  int* idx0 = allocI(32 * 256 * 32);
  int* idx1 = allocI(32 * 64 * 32);
  int* idx2 = allocI(32 * 16 * 32);
  float* c00 = allocF(32 * 256 * 6);
  float* c10 = allocF(32 * 256 * 3 * 7);
  float* c20 = allocF(32 * 256 * 5 * 3);
  float* c30 = allocF(32 * 256 * 7 * 3);
  float* n0b = allocF(32 * 256 * 19);
  float* y00 = allocF(32 * 256 * 32);
  float* y10 = allocF(32 * 256 * 3 * 32);
  float* y20 = allocF(32 * 256 * 5 * 32);
  float* y30 = allocF(32 * 256 * 7 * 32);
  float* gz  = allocF(32 * 256 * 32);
  float* c01 = allocF(32 * 64 * 387);
  float* c11 = allocF(32 * 64 * 3 * 196);
  float* c21 = allocF(32 * 64 * 5 * 192);
  float* c31 = allocF(32 * 64 * 7 * 192);
  float* n1b = allocF(32 * 64 * 967);
  float* y01 = allocF(32 * 64 * 64);
  float* y11 = allocF(32 * 64 * 3 * 64);
  float* y21 = allocF(32 * 64 * 5 * 64);
  float* y31 = allocF(32 * 64 * 7 * 64);
  float* c02 = allocF(32 * 16 * 771);
  float* c12 = allocF(32 * 16 * 3 * 388);
  float* c22 = allocF(32 * 16 * 5 * 384);
  float* c32 = allocF(32 * 16 * 7 * 384);
  float* n2b = allocF(32 * 16 * 1927);
  float* y02 = allocF(32 * 16 * 128);
  float* y12 = allocF(32 * 16 * 3 * 128);
  float* y22 = allocF(32 * 16 * 5 * 128);
  float* y32 = allocF(32 * 16 * 7 * 128);
  float* n3b = allocF(32 * 16 * 512);
  float* fbuf = allocF(32 * 512);
  float* h1 = allocF(32 * 512);
  float* h2 = allocF(32 * 256);
  float* lg = allocF(32 * 40);

  auto egrid = [](long n) { return (unsigned)((n + 255) / 256); };
  auto gemm = [&](const float* A, const float* B, float* C, int M, int K, int N) {
    int tilesM = M / 16;
    if (N % 64 == 0) {
      gemm_wmma_kernel<4, false><<<tilesM * (N / 64), 32, 0, stream>>>(A, B, C, M, K, N);
    } else if (N % 32 == 0) {
      gemm_wmma_kernel<2, false><<<tilesM * (N / 32), 32, 0, stream>>>(A, B, C, M, K, N);
    } else {
      int tnw = (N + 63) / 64;
      gemm_wmma_kernel<4, true><<<tilesM * tnw, 32, 0, stream>>>(A, B, C, M, K, N);
    }
  };

  // 1) kd-tree indexing: 10 bitonic-sort levels, ping-pong buffers
  const float* cur = X;
  float* bufs[2] = {xsB, xsA};
  int flip = 0;
  for (int lev = 0; lev < 10; lev++) {
    int nb = 1 << lev, m = 1024 >> lev;
    kd_sort_kernel<<<32 * nb, 256, 0, stream>>>(cur, bufs[flip], m);
    cur = bufs[flip];
    flip ^= 1;
  }
  const float* pts0 = cur;

  // 2) kd pooling
  kd_pool_kernel<<<egrid(32 * 256 * 3), 256, 0, stream>>>(pts0, pts1, 256, 4);
  kd_pool_kernel<<<egrid(32 * 64 * 3), 256, 0, stream>>>(pts1, pts2, 64, 4);
  kd_pool_kernel<<<egrid(32 * 16 * 3), 256, 0, stream>>>(pts2, pts3, 16, 4);

  // ---------------- level 0 ----------------
  knn_kernel<<<32 * 256, 32, 0, stream>>>(pts0, pts1, idx0, 1024, 256);
  sh_conv_kernel<0><<<32 * 256, 256, 0, stream>>>(pts0, pts1, idx0,
      nullptr, nullptr, nullptr, nullptr, c00, c10, c20, c30);
  inv_norm_kernel<<<egrid(8192 * 6), 256, 0, stream>>>(c00, n0b, 8192, 1, 6, 19, 0);
  inv_norm_kernel<<<egrid(8192 * 7), 256, 0, stream>>>(c10, n0b, 8192, 3, 7, 19, 6);
  inv_norm_kernel<<<egrid(8192 * 3), 256, 0, stream>>>(c20, n0b, 8192, 5, 3, 19, 13);
  inv_norm_kernel<<<egrid(8192 * 3), 256, 0, stream>>>(c30, n0b, 8192, 7, 3, 19, 16);
  gemm(c00, EQW(0, 0), y00, 8192, 6, 32);
  relu_kernel<<<egrid(8192 * 32), 256, 0, stream>>>(y00, 8192 * 32);
  gemm(c10, EQW(0, 1), y10, 24576, 7, 32);
  gemm(n0b, GW(0, 0), gz, 8192, 19, 32);
  gate_apply_kernel<<<egrid(24576L * 32), 256, 0, stream>>>(y10, gz, GB(0, 0), 8192, 3, 32);
  gemm(c20, EQW(0, 2), y20, 40960, 3, 32);
  gemm(n0b, GW(0, 1), gz, 8192, 19, 32);
  gate_apply_kernel<<<egrid(40960L * 32), 256, 0, stream>>>(y20, gz, GB(0, 1), 8192, 5, 32);
  gemm(c30, EQW(0, 3), y30, 57344, 3, 32);
  gemm(n0b, GW(0, 2), gz, 8192, 19, 32);
  gate_apply_kernel<<<egrid(57344L * 32), 256, 0, stream>>>(y30, gz, GB(0, 2), 8192, 7, 32);

  // ---------------- level 1 ----------------
  knn_kernel<<<32 * 64, 32, 0, stream>>>(pts1, pts2, idx1, 256, 64);
  sh_conv_kernel<1><<<32 * 64, 256, 0, stream>>>(pts1, pts2, idx1,
      y00, y10, y20, y30, c01, c11, c21, c31);
  inv_norm_kernel<<<egrid(2048 * 387), 256, 0, stream>>>(c01, n1b, 2048, 1, 387, 967, 0);
  inv_norm_kernel<<<egrid(2048 * 196), 256, 0, stream>>>(c11, n1b, 2048, 3, 196, 967, 387);
  inv_norm_kernel<<<egrid(2048 * 192), 256, 0, stream>>>(c21, n1b, 2048, 5, 192, 967, 583);
  inv_norm_kernel<<<egrid(2048 * 192), 256, 0, stream>>>(c31, n1b, 2048, 7, 192, 967, 775);
  gemm(c01, EQW(1, 0), y01, 2048, 387, 64);
  relu_kernel<<<egrid(2048 * 64), 256, 0, stream>>>(y01, 2048 * 64);
  gemm(c11, EQW(1, 1), y11, 6144, 196, 64);
  gemm(n1b, GW(1, 0), gz, 2048, 967, 64);
  gate_apply_kernel<<<egrid(6144L * 64), 256, 0, stream>>>(y11, gz, GB(1, 0), 2048, 3, 64);
  gemm(c21, EQW(1, 2), y21, 10240, 192, 64);
  gemm(n1b, GW(1, 1), gz, 2048, 967, 64);
  gate_apply_kernel<<<egrid(10240L * 64), 256, 0, stream>>>(y21, gz, GB(1, 1), 2048, 5, 64);
  gemm(c31, EQW(1, 3), y31, 14336, 192, 64);
  gemm(n1b, GW(1, 2), gz, 2048, 967, 64);
  gate_apply_kernel<<<egrid(14336L * 64), 256, 0, stream>>>(y31, gz, GB(1, 2), 2048, 7, 64);

  // ---------------- level 2 ----------------
  knn_kernel<<<32 * 16, 32, 0, stream>>>(pts2, pts3, idx2, 64, 16);
  sh_conv_kernel<2><<<32 * 16, 256, 0, stream>>>(pts2, pts3, idx2,
      y01, y11, y21, y31, c02, c12, c22, c32);
  inv_norm_kernel<<<egrid(512 * 771), 256, 0, stream>>>(c02, n2b, 512, 1, 771, 1927, 0);
  inv_norm_kernel<<<egrid(512 * 388), 256, 0, stream>>>(c12, n2b, 512, 3, 388, 1927, 771);
  inv_norm_kernel<<<egrid(512 * 384), 256, 0, stream>>>(c22, n2b, 512, 5, 384, 1927, 1159);
  inv_norm_kernel<<<egrid(512 * 384), 256, 0, stream>>>(c32, n2b, 512, 7, 384, 1927, 1543);
  gemm(c02, EQW(2, 0), y02, 512, 771, 128);
  relu_kernel<<<egrid(512 * 128), 256, 0, stream>>>(y02, 512 * 128);
  gemm(c12, EQW(2, 1), y12, 1536, 388, 128);
  gemm(n2b, GW(2, 0), gz, 512, 1927, 128);
  gate_apply_kernel<<<egrid(1536L * 128), 256, 0, stream>>>(y12, gz, GB(2, 0), 512, 3, 128);
  gemm(c22, EQW(2, 2), y22, 2560, 384, 128);
  gemm(n2b, GW(2, 1), gz, 512, 1927, 128);
  gate_apply_kernel<<<egrid(2560L * 128), 256, 0, stream>>>(y22, gz, GB(2, 1), 512, 5, 128);
  gemm(c32, EQW(2, 3), y32, 3584, 384, 128);
  gemm(n2b, GW(2, 2), gz, 512, 1927, 128);
  gate_apply_kernel<<<egrid(3584L * 128), 256, 0, stream>>>(y32, gz, GB(2, 2), 512, 7, 128);

  // ---------------- head ----------------
  inv_norm_kernel<<<egrid(512 * 128), 256, 0, stream>>>(y02, n3b, 512, 1, 128, 512, 0);
  inv_norm_kernel<<<egrid(512 * 128), 256, 0, stream>>>(y12, n3b, 512, 3, 128, 512, 128);
  inv_norm_kernel<<<egrid(512 * 128), 256, 0, stream>>>(y22, n3b, 512, 5, 128, 512, 256);
  inv_norm_kernel<<<egrid(512 * 128), 256, 0, stream>>>(y32, n3b, 512, 7, 128, 512, 384);
  maxpool_kernel<<<egrid(32 * 512), 256, 0, stream>>>(n3b, fbuf);

  gemm(fbuf, fc1w, h1, 32, 512, 512);
  bias_act_kernel<<<egrid(32 * 512), 256, 0, stream>>>(h1, fc1b, 32, 512, 1);
  gemm(h1, fc2w, h2, 32, 512, 256);
  bias_act_kernel<<<egrid(32 * 256), 256, 0, stream>>>(h2, fc2b, 32, 256, 1);
  gemm(h2, outw, lg, 32, 256, 40);
  softmax_kernel<<<1, 32, 0, stream>>>(lg, outb, OUT);
}